// LogScale_40776419508797
// MI455X (gfx1250) — compile-verified
//
#include <hip/hip_runtime.h>
#include <hip/hip_bf16.h>
#include <stdint.h>

#define N_IN      2049
#define N_OUT     1024
#define BLOCK     256
#define R         4        // rows per block (table reuse factor)
#define ROWSTRIDE 2052     // 2049 + 3 pad DWORDs -> LDS bank shift of 4 per row

typedef unsigned int u32x4 __attribute__((ext_vector_type(4)));
typedef int          i32x8 __attribute__((ext_vector_type(8)));
typedef int          i32x4 __attribute__((ext_vector_type(4)));

// R rows per workgroup. Waves 0..R-1 each issue one Tensor-DMA descriptor
// loading their row into an LDS slot (per-wave TENSORcnt), then all 1024
// scattered gathers are served from LDS, with every constant-table element
// read once and applied to R rows.
__global__ __launch_bounds__(BLOCK) void LogScale_kernel(
    const float* __restrict__ x,          // (rows, 2049)
    const float* __restrict__ frac_lin,   // (n_lin)
    const float* __restrict__ cubic_t,    // (n_cub)
    const float* __restrict__ tri_w,      // (n_tri, max_w)
    const int*   __restrict__ pair_idx,   // (2*n_lin)
    const int*   __restrict__ cubic_i0,   // (n_cub)
    const int*   __restrict__ tri_idx,    // (n_tri, max_w)
    float*       __restrict__ out,        // (rows, 1024)
    int n_lin, int n_cub, int max_w, int n_rows)
{
    __shared__ float rows[R * ROWSTRIDE];             // 4 * 8208 B = 32.8 KB

    const int row0 = blockIdx.x * R;
    const int nr   = (n_rows - row0 < R) ? (n_rows - row0) : R;

#if __has_builtin(__builtin_amdgcn_tensor_load_to_lds)
    {
        const int wave = threadIdx.x >> 5;
        if (((threadIdx.x & 31) == 0) && (wave < nr)) {
            // ---- Tensor DMA descriptor (D#): 1-D tile of 2049 f32 ----
            const unsigned lds_off =
                (unsigned)(unsigned long long)(const void*)&rows[0]
                + (unsigned)(wave * ROWSTRIDE * 4);
            const unsigned long long ga =
                (unsigned long long)(const void*)(x + (size_t)(row0 + wave) * N_IN);

            u32x4 g0;
            g0.x = 1u;                                        // count=1, user descriptor
            g0.y = lds_off;                                   // lds_addr (bytes)
            g0.z = (unsigned)(ga & 0xFFFFFFFFull);            // global_addr[31:0]
            g0.w = (unsigned)((ga >> 32) & 0x01FFFFFFull)     // global_addr[56:32]
                 | 0x80000000u;                               // type = 2 ("image")

            i32x8 g1;
            g1[0] = (2 << 16);                                // data_size = 4 bytes
            g1[1] = (int)((unsigned)(N_IN & 0xFFFF) << 16);   // tensor_dim0[15:0] @ 63:48
            g1[2] = 0;                                        // dim0 hi / dim1 lo
            g1[3] = (int)((unsigned)(N_IN & 0xFFFF) << 16);   // tile_dim0 @ 127:112
            g1[4] = 0;                                        // tile_dim1 = 0 (1-D tile)
            g1[5] = N_IN;                                     // tensor_dim0_stride
            g1[6] = 0;
            g1[7] = 0;

            i32x4 gz = {0, 0, 0, 0};
#if defined(__clang_major__) && (__clang_major__ >= 23)
            i32x8 gz8 = {0, 0, 0, 0, 0, 0, 0, 0};
            __builtin_amdgcn_tensor_load_to_lds(g0, g1, gz, gz, gz8, 0);
#else
            __builtin_amdgcn_tensor_load_to_lds(g0, g1, gz, gz, 0);
#endif
            __builtin_amdgcn_s_wait_tensorcnt(0);             // this wave's DMA done
        }
    }
#else
    // Fallback: cooperative coalesced fill.
    for (int i = threadIdx.x; i < nr * N_IN; i += BLOCK) {
        const int r = i / N_IN, c = i - r * N_IN;
        rows[r * ROWSTRIDE + c] = x[(size_t)(row0 + r) * N_IN + c];
    }
#endif
    __syncthreads();

    const int n_lc = n_lin + n_cub;

    #pragma unroll
    for (int jj = 0; jj < N_OUT / BLOCK; ++jj) {
        const int j = jj * BLOCK + threadIdx.x;
        float v[R];

        if (j < n_lin) {
            // ---- linear interpolation: 1 table read -> R rows ----
            const int   i0 = pair_idx[j];
            const int   i1 = pair_idx[n_lin + j];
            const float f  = frac_lin[j];
            #pragma unroll
            for (int r = 0; r < R; ++r) {
                const float x0 = rows[r * ROWSTRIDE + i0];
                const float x1 = rows[r * ROWSTRIDE + i1];
                v[r] = x0 + f * (x1 - x0);
            }
        } else if (j < n_lc) {
            // ---- Catmull-Rom cubic ----
            const int   k  = j - n_lin;
            const int   i0 = cubic_i0[k];
            const float t  = cubic_t[k];
            #pragma unroll
            for (int r = 0; r < R; ++r) {
                const float xm1 = rows[r * ROWSTRIDE + i0 - 1];
                const float x0  = rows[r * ROWSTRIDE + i0];
                const float x1  = rows[r * ROWSTRIDE + i0 + 1];
                const float x2  = rows[r * ROWSTRIDE + i0 + 2];
                v[r] = x0 + 0.5f * t * (x1 - xm1
                       + t * (2.0f * xm1 - 5.0f * x0 + 4.0f * x1 - x2
                       + t * (3.0f * (x0 - x1) + x2 - xm1)));
            }
        } else {
            // ---- triangular window: each (idx, w) table pair read once,
            //      applied to R rows (bank-shifted LDS gathers) ----
            const int k = j - n_lc;
            const int*   __restrict__ ip = tri_idx + (size_t)k * max_w;
            const float* __restrict__ wp = tri_w   + (size_t)k * max_w;
            #pragma unroll
            for (int r = 0; r < R; ++r) v[r] = -__builtin_inff();
            for (int w = 0; w < max_w; ++w) {
                const int   idx = ip[w];
                const float wgt = wp[w];
                #pragma unroll
                for (int r = 0; r < R; ++r) {
                    v[r] = fmaxf(v[r], rows[r * ROWSTRIDE + idx] + wgt);
                }
            }
        }

        #pragma unroll
        for (int r = 0; r < R; ++r) {
            if (r < nr) out[(size_t)(row0 + r) * N_OUT + j] = v[r];
        }
    }
}

extern "C" void kernel_launch(void* const* d_in, const int* in_sizes, int n_in,
                              void* d_out, int out_size, void* d_ws, size_t ws_size,
                              hipStream_t stream) {
    // setup_inputs order:
    //   0: x (B,F,2049) f32       1: fraction_linear (n_lin) f32
    //   2: cubic_t (n_cub) f32    3: triangular_weights (n_tri,max_w) f32
    //   4: linear_pair_idx (2*n_lin) i32   5: cubic_i0 (n_cub) i32
    //   6: triangular_idx (n_tri,max_w) i32
    const float* x    = (const float*)d_in[0];
    const float* fl   = (const float*)d_in[1];
    const float* ct   = (const float*)d_in[2];
    const float* tw   = (const float*)d_in[3];
    const int*   pair = (const int*)  d_in[4];
    const int*   ci0  = (const int*)  d_in[5];
    const int*   tidx = (const int*)  d_in[6];

    const int n_lin  = in_sizes[1];
    const int n_cub  = in_sizes[2];
    const int n_tri  = N_OUT - n_lin - n_cub;
    const int max_w  = (n_tri > 0) ? (in_sizes[3] / n_tri) : 0;
    const int n_rows = in_sizes[0] / N_IN;          // B*F = 1600
    const int blocks = (n_rows + R - 1) / R;        // 400

    LogScale_kernel<<<blocks, BLOCK, 0, stream>>>(
        x, fl, ct, tw, pair, ci0, tidx, (float*)d_out,
        n_lin, n_cub, max_w, n_rows);
}